// GCM_Multi_Basics_80788334838514
// MI455X (gfx1250) — compile-verified
//
#include <hip/hip_runtime.h>

typedef float v2f __attribute__((ext_vector_type(2)));
typedef float v8f __attribute__((ext_vector_type(8)));
typedef unsigned int v4u __attribute__((ext_vector_type(4)));
typedef int v4i __attribute__((ext_vector_type(4)));
typedef int v8i __attribute__((ext_vector_type(8)));
typedef __attribute__((address_space(3))) float lds_f;

#define IMG       512
#define KS        21
#define KK        441          // 21*21
#define KPAD      444          // pad to multiple of 4 (WMMA K-step)
#define NKSTEP    111          // 444 / 4
#define MROWS     22           // effective kernels per (b,c)
#define TILE_ROWS 8
#define ROWS_PP   4            // rows accumulated per K-sweep (A-frag reuse)
#define TILE_COLS 64           // 4 waves * 16 pixels
#define HALO      10
#define XS_H      (TILE_ROWS + KS)       // 29: padded taps (dy=21) stay in-bounds
#define XS_W      (TILE_COLS + 2*HALO)   // 84

// ---------------------------------------------------------------------------
// Kernel 1: build per-(b,c) mixed kernel matrices A[24][22][444]
//   row r of pair (b,c) = sum_g weight[b,g,k] * base_weight[g,k],
//   k = (22*c + r)/3, 441 taps zero-padded to 444 along K.
// ---------------------------------------------------------------------------
__global__ void build_mixed_kernels(const float* __restrict__ w,    // [8,4,22]
                                    const float* __restrict__ bw,   // [4,22,441]
                                    float* __restrict__ A, int total) {
  int i = blockIdx.x * blockDim.x + threadIdx.x;
  if (i >= total) return;
  int t   = i % KPAD;
  int rem = i / KPAD;
  int r   = rem % MROWS;
  int bc  = rem / MROWS;
  int b = bc / 3, c = bc % 3;
  float v = 0.f;
  if (t < KK) {
    int k = (MROWS * c + r) / 3;
#pragma unroll
    for (int g = 0; g < 4; ++g)
      v += w[(b * 4 + g) * MROWS + k] * bw[(g * MROWS + k) * KK + t];
  }
  A[i] = v;
}

// ---------------------------------------------------------------------------
// Kernel 2: implicit-GEMM 21x21 conv via V_WMMA_F32_16X16X4_F32.
//   M = 22 kernels (two 16-row tiles), N = 16 pixels/wave, K = 444 by 4.
//   K-sweep accumulates 4 output rows at once so A fragments are reused 4x.
// ---------------------------------------------------------------------------
__global__ void __launch_bounds__(128)
gcm_conv_wmma(const float* __restrict__ x,   // [8,3,512,512]
              const float* __restrict__ A,   // [24,22,444]
              float* __restrict__ out) {     // [8,66,512,512]
  __shared__ float As[MROWS * KPAD];         // 39,072 B
  __shared__ float Xs[XS_H * XS_W];          //  9,744 B

  const int z  = blockIdx.z;                 // b*3 + c
  const int b  = z / 3, c = z % 3;
  const int y0 = blockIdx.y * TILE_ROWS;
  const int x0 = blockIdx.x * TILE_COLS;
  const int tid  = threadIdx.x;
  const int wave = tid >> 5;
  const int lane = tid & 31;

  const float* Az = A + (size_t)z * (MROWS * KPAD);

  // --- stage mixed kernel matrix: Tensor Data Mover (L2 -> LDS DMA) ---
#if __has_builtin(__builtin_amdgcn_tensor_load_to_lds)
  if (tid < 32) {
    const unsigned long long ga = (unsigned long long)(const void*)Az;
    const lds_f* as3 = (const lds_f*)&As[0];
    const unsigned lds_off = (unsigned)(unsigned long long)as3;
    const unsigned nelem = MROWS * KPAD;      // 9768 4-byte elements
    v4u g0; v8i g1;
    v4i gz = {0, 0, 0, 0};
    g0[0] = 1u;                                            // count=1 (valid D#)
    g0[1] = lds_off;                                       // lds_addr (bytes)
    g0[2] = (unsigned)(ga & 0xFFFFFFFFull);                // global_addr[31:0]
    g0[3] = (unsigned)((ga >> 32) & 0x01FFFFFFull)         // global_addr[56:32]
            | (2u << 30);                                  // type=2 ("image")
    g1[0] = (int)(2u << 16);                               // data_size=4B
    g1[1] = (int)((nelem & 0xFFFFu) << 16);                // tensor_dim0 lo16 @48
    g1[2] = (int)((nelem >> 16) | (1u << 16));             // dim0 hi16 | tensor_dim1=1
    g1[3] = (int)((nelem & 0xFFFFu) << 16);                // tile_dim0 @127:112
    g1[4] = 1;                                             // tile_dim1=1
    g1[5] = (int)nelem;                                    // tensor_dim0_stride
    g1[6] = 0; g1[7] = 0;
#if defined(__clang_major__) && (__clang_major__ >= 23)
    v8i g4 = {0, 0, 0, 0, 0, 0, 0, 0};
    __builtin_amdgcn_tensor_load_to_lds(g0, g1, gz, gz, g4, 0);
#else
    __builtin_amdgcn_tensor_load_to_lds(g0, g1, gz, gz, 0);
#endif
  }
#else
  for (int i = tid; i < MROWS * KPAD; i += 128) As[i] = Az[i];
#endif

  // --- stage haloed input tile (manual, overlaps the TDM DMA) ---
  const float* xz = x + (size_t)z * (IMG * IMG);
  for (int i = tid; i < XS_H * XS_W; i += 128) {
    int ry = i / XS_W, cx = i - ry * XS_W;
    int gy = y0 + ry - HALO, gx = x0 + cx - HALO;
    float v = 0.f;
    if (gy >= 0 && gy < IMG && gx >= 0 && gx < IMG) {
      v = xz[gy * IMG + gx];
      if (gy + TILE_ROWS < IMG)               // hint next row-tile into caches
        __builtin_prefetch(xz + (gy + TILE_ROWS) * IMG + gx, 0, 0);
    }
    Xs[i] = v;
  }
#if __has_builtin(__builtin_amdgcn_tensor_load_to_lds)
  __builtin_amdgcn_s_wait_tensorcnt(0);       // wave0's DMA complete
#endif
  __syncthreads();

  const int ln   = lane & 15;        // N-col / M-row selector
  const int kh   = lane >> 4;        // K-half: K = 4*ks + 2*kh + v
  const int xoff = wave * 16 + ln;
  const int r1   = 16 + ln;          // M-tile-1 row
  const bool r1ok = (r1 < MROWS);
  const int r1c  = r1ok ? r1 : (MROWS - 1);

  for (int rp = 0; rp < TILE_ROWS; rp += ROWS_PP) {
    v8f zero = {0.f, 0.f, 0.f, 0.f, 0.f, 0.f, 0.f, 0.f};
    v8f acc[ROWS_PP][2];
#pragma unroll
    for (int rr = 0; rr < ROWS_PP; ++rr) { acc[rr][0] = zero; acc[rr][1] = zero; }

    for (int ks = 0; ks < NKSTEP; ++ks) {
      v2f af0, af1;
      int toff[2];
#pragma unroll
      for (int v = 0; v < 2; ++v) {
        const int t  = ks * 4 + 2 * kh + v;    // tap index, < 444
        const int dy = t / KS;
        const int dx = t - dy * KS;
        af0[v] = As[ln * KPAD + t];                    // pad cols are zero
        af1[v] = r1ok ? As[r1c * KPAD + t] : 0.f;      // rows >=22 are zero
        toff[v] = (rp + dy) * XS_W + xoff + dx;        // in-bounds even at pad
      }
#pragma unroll
      for (int rr = 0; rr < ROWS_PP; ++rr) {           // reuse A frags 4x
        v2f bf;
        bf[0] = Xs[toff[0] + rr * XS_W];
        bf[1] = Xs[toff[1] + rr * XS_W];
        acc[rr][0] = __builtin_amdgcn_wmma_f32_16x16x4_f32(
            false, af0, false, bf, (short)0, acc[rr][0], false, false);
        acc[rr][1] = __builtin_amdgcn_wmma_f32_16x16x4_f32(
            false, af1, false, bf, (short)0, acc[rr][1], false, false);
      }
    }

    // --- store: out channel j = 22*c + m, m = Mtile*16 + v + 8*kh ---
#pragma unroll
    for (int rr = 0; rr < ROWS_PP; ++rr) {
      const int y  = y0 + rp + rr;
      const int px = x0 + wave * 16 + ln;
      const size_t obase = (((size_t)b * 66 + 22 * c) * IMG + y) * IMG + px;
#pragma unroll
      for (int v = 0; v < 8; ++v) {
        const int m0 = v + 8 * kh;             // 0..15
        out[obase + (size_t)m0 * IMG * IMG] = acc[rr][0][v];
        const int m1 = 16 + m0;                // valid only < 22
        if (m1 < MROWS)
          out[obase + (size_t)m1 * IMG * IMG] = acc[rr][1][v];
      }
    }
  }
}

// ---------------------------------------------------------------------------
extern "C" void kernel_launch(void* const* d_in, const int* in_sizes, int n_in,
                              void* d_out, int out_size, void* d_ws, size_t ws_size,
                              hipStream_t stream) {
  (void)in_sizes; (void)n_in; (void)out_size; (void)ws_size;
  const float* x  = (const float*)d_in[0];   // [8,3,512,512]
  const float* w  = (const float*)d_in[1];   // [8,4,22,1,1,1]
  const float* bw = (const float*)d_in[2];   // [4,22,1,21,21]
  float* out = (float*)d_out;                // [8,66,512,512]
  float* A   = (float*)d_ws;                 // [24,22,444] = 0.94 MB

  const int totalA = 24 * MROWS * KPAD;
  build_mixed_kernels<<<(totalA + 255) / 256, 256, 0, stream>>>(w, bw, A, totalA);

  dim3 grid(IMG / TILE_COLS, IMG / TILE_ROWS, 24);  // (8, 64, 24)
  gcm_conv_wmma<<<grid, dim3(128), 0, stream>>>(x, A, out);
}